// GraphConvolution_35158602285612
// MI455X (gfx1250) — compile-verified
//
#include <hip/hip_runtime.h>

// GCN layer for MI455X (gfx1250, wave32):
//   support = X @ W          -- dense GEMM via V_WMMA_F32_16X16X4_F32 (native f32 WMMA)
//   out     = A_sparse @ support + bias   -- edge-parallel scatter with global_atomic_add_f32
//
// N=50000, E=800000, IN=256, OUT=128. All tile dims divide exactly:
//   50000/16 = 3125 row tiles, 128/16 = 8 col tiles -> 25000 tiles, 8 waves/block.

typedef float v2f __attribute__((ext_vector_type(2)));
typedef float v8f __attribute__((ext_vector_type(8)));

#define GCN_N    50000
#define GCN_E    800000
#define GCN_IN   256
#define GCN_OUT  128

// ---------------------------------------------------------------------------
// Kernel 1: support[50000,128] = X[50000,256] @ W[256,128]
// One wave32 computes one 16x16 tile; K-loop steps 4 with f32 WMMA.
//
// A (16x4 f32, 2 VGPRs):  lanes 0-15: M=lane, K={0,1}; lanes 16-31: M=lane-16, K={2,3}
// B (4x16 f32, 2 VGPRs):  lanes 0-15: N=lane, K={0,1}; lanes 16-31: N=lane-16, K={2,3}
// C/D (16x16 f32, 8 VGPRs): VGPR r: lanes 0-15 -> M=r, N=lane; lanes 16-31 -> M=r+8, N=lane-16
// ---------------------------------------------------------------------------
__global__ __launch_bounds__(256) void gcn_gemm_wmma(const float* __restrict__ X,
                                                     const float* __restrict__ W,
                                                     float* __restrict__ support) {
    const int lane  = threadIdx.x & 31;
    const int wave  = threadIdx.x >> 5;
    const int tile  = blockIdx.x * 8 + wave;          // 25000 tiles total
    const int row0  = (tile >> 3) << 4;               // 0..49984 step 16
    const int col0  = (tile & 7) << 4;                // 0..112 step 16
    const int m     = lane & 15;                      // M (for A) == N (for B)
    const int khalf = lane >> 4;                      // 0: K={0,1}; 1: K={2,3}

    const float* __restrict__ xrow = X + (size_t)(row0 + m) * GCN_IN;
    const float* __restrict__ wcol = W + (size_t)col0 + m;

    v8f c = {};
#pragma unroll 8
    for (int k0 = 0; k0 < GCN_IN; k0 += 4) {
        const int kk = k0 + khalf * 2;
        // A: two consecutive K values of this row -> one 8-byte load (kk is even)
        v2f a = *(const v2f*)(xrow + kk);
        // B: two consecutive K rows of W at column (col0+m)
        v2f b;
        b.x = wcol[(size_t)kk * GCN_OUT];
        b.y = wcol[(size_t)(kk + 1) * GCN_OUT];
        // D = A*B + C    (8 args: neg_a, A, neg_b, B, c_mod, C, reuse_a, reuse_b)
        c = __builtin_amdgcn_wmma_f32_16x16x4_f32(false, a, false, b,
                                                  (short)0, c, false, false);
    }

    // Store 16x16 tile per the C/D layout.
    float* __restrict__ orow = support + (size_t)(row0 + khalf * 8) * GCN_OUT + col0 + m;
#pragma unroll
    for (int r = 0; r < 8; ++r) {
        orow[(size_t)r * GCN_OUT] = c[r];
    }
}

// ---------------------------------------------------------------------------
// Kernel 2: out[n][j] = bias[j]   (also clears the 0xAA poison in d_out)
// One float4 per thread. 50000*32 float4s = 6250 blocks of 256.
// ---------------------------------------------------------------------------
__global__ __launch_bounds__(256) void gcn_init(const float4* __restrict__ bias4,
                                                float4* __restrict__ out4) {
    const unsigned idx = blockIdx.x * 256u + threadIdx.x;   // < N*32
    out4[idx] = bias4[idx & 31];                             // 128/4 = 32 float4s per row
}

// ---------------------------------------------------------------------------
// Kernel 3: for each edge e: out[rows[e]] += vals[e] * support[cols[e]]
// 32 lanes per edge; each lane owns a contiguous float4 of the 128-wide row.
// support & out live in L2 (25.6 MB each << 192 MB), so gathers and the FP32
// atomics are L2-resident. unsafeAtomicAdd -> global_atomic_add_f32.
// ---------------------------------------------------------------------------
__global__ __launch_bounds__(256) void gcn_spmm(const int* __restrict__ rows,
                                                const int* __restrict__ cols,
                                                const float* __restrict__ vals,
                                                const float* __restrict__ support,
                                                float* __restrict__ out) {
    const unsigned gid   = blockIdx.x * 256u + threadIdx.x;
    const unsigned e     = gid >> 5;          // edge index (exactly E*32 threads launched)
    const unsigned chunk = (gid & 31) << 2;   // float offset within the 128-wide row

    const int   r    = rows[e];
    const int   cidx = cols[e];
    const float v    = vals[e];

    const float4 s = *(const float4*)(support + (size_t)cidx * GCN_OUT + chunk);
    float* __restrict__ o = out + (size_t)r * GCN_OUT + chunk;

    unsafeAtomicAdd(o + 0, v * s.x);
    unsafeAtomicAdd(o + 1, v * s.y);
    unsafeAtomicAdd(o + 2, v * s.z);
    unsafeAtomicAdd(o + 3, v * s.w);
}

// ---------------------------------------------------------------------------
extern "C" void kernel_launch(void* const* d_in, const int* in_sizes, int n_in,
                              void* d_out, int out_size, void* d_ws, size_t ws_size,
                              hipStream_t stream) {
    const int*   adj_rows = (const int*)d_in[0];
    const int*   adj_cols = (const int*)d_in[1];
    const float* adj_vals = (const float*)d_in[2];
    const float* X        = (const float*)d_in[3];   // [50000, 256]
    const float* W        = (const float*)d_in[4];   // [256, 128]
    const float* bias     = (const float*)d_in[5];   // [128]

    float* out     = (float*)d_out;                  // [50000, 128]
    float* support = (float*)d_ws;                   // [50000, 128] scratch (25.6 MB)

    // GEMM: 25000 tiles, 8 waves (one tile each) per block.
    gcn_gemm_wmma<<<3125, 256, 0, stream>>>(X, W, support);

    // out = broadcast(bias)
    gcn_init<<<(GCN_N * 32) / 256, 256, 0, stream>>>((const float4*)bias, (float4*)out);

    // Scatter-add over edges (depends on both kernels above; same stream orders them).
    gcn_spmm<<<(GCN_E * 32) / 256, 256, 0, stream>>>(adj_rows, adj_cols, adj_vals,
                                                     support, out);
}